// SDFModule_14233521619468
// MI455X (gfx1250) — compile-verified
//
#include <hip/hip_runtime.h>
#include <hip/hip_bf16.h>
#include <stdint.h>

#define THREADS 256
#define TILE    512                 // triangles per LDS tile
#define F4_PER_TILE (TILE * 3)      // 3 float4 per triangle: (a,nab) (ab,g) (ac,nac)
#define COPIES  (F4_PER_TILE / THREADS)   // 6 b128 async copies per thread per tile

#define EPS_F      1e-12f
#define MAX_DIST_F 1e8f
#define INV_TWO_PI 0.15915494309189535f

// ---------- CDNA5 feature detection ----------
#if defined(__has_builtin)
#  if __has_builtin(__builtin_amdgcn_global_load_async_to_lds_b128)
#    define HAS_ASYNC_LDS 1
#  else
#    define HAS_ASYNC_LDS 0
#  endif
#  if __has_builtin(__builtin_amdgcn_s_wait_asynccnt)
#    define HAS_WAIT_ASYNC_BUILTIN 1
#  else
#    define HAS_WAIT_ASYNC_BUILTIN 0
#  endif
#else
#  define HAS_ASYNC_LDS 0
#  define HAS_WAIT_ASYNC_BUILTIN 0
#endif

#if HAS_ASYNC_LDS
#  if HAS_WAIT_ASYNC_BUILTIN
#    define WAIT_ASYNC(n) __builtin_amdgcn_s_wait_asynccnt(n)
#  else
#    define WAIT_ASYNC(n) asm volatile("s_wait_asynccnt %0" :: "n"(n) : "memory")
#  endif
#else
#  define WAIT_ASYNC(n) ((void)0)
#endif

// typed pointers matching the async-LDS builtin's signature
typedef int v4i __attribute__((ext_vector_type(4)));
typedef __attribute__((address_space(1))) v4i* gptr_v4i;
typedef __attribute__((address_space(3))) v4i* lptr_v4i;

// ---------- fast math helpers ----------
__device__ __forceinline__ float frcp(float x) {
#if defined(__has_builtin) && __has_builtin(__builtin_amdgcn_rcpf)
    return __builtin_amdgcn_rcpf(x);
#else
    return 1.0f / x;
#endif
}
__device__ __forceinline__ float fsqrt(float x) {
#if defined(__has_builtin) && __has_builtin(__builtin_amdgcn_sqrtf)
    return __builtin_amdgcn_sqrtf(x);
#else
    return sqrtf(x);
#endif
}
__device__ __forceinline__ float clamp01(float x) {
    return fminf(fmaxf(x, 0.0f), 1.0f);
}
__device__ __forceinline__ float dot3(float ax, float ay, float az,
                                      float bx, float by, float bz) {
    return fmaf(ax, bx, fmaf(ay, by, az * bz));
}

// minimax atan on [0,1], max err ~2e-6; full-quadrant atan2
__device__ __forceinline__ float fast_atan2f(float y, float x) {
    float ax = fabsf(x), ay = fabsf(y);
    float mx = fmaxf(ax, ay);
    float mn = fminf(ax, ay);
    float z  = mn * frcp(fmaxf(mx, 1e-38f));
    float z2 = z * z;
    float p  = fmaf(z2, -0.0117212f, 0.05265332f);
    p = fmaf(z2, p, -0.11643287f);
    p = fmaf(z2, p,  0.19354346f);
    p = fmaf(z2, p, -0.33262347f);
    p = fmaf(z2, p,  0.99997726f);
    float r = p * z;
    r = (ay > ax)   ? (1.57079632679f - r) : r;
    r = (x < 0.0f)  ? (3.14159265359f - r) : r;
    return __builtin_copysignf(r, y);
}

// ---------- per point x triangle body ----------
// T0 = (a.xyz, |ab|^2), T1 = (ab.xyz, ab.ac), T2 = (ac.xyz, |ac|^2)
__device__ __forceinline__ void pair_accum(float px, float py, float pz,
                                           float4 T0, float4 T1, float4 T2,
                                           float& minsq, float& wsum) {
    float abx = T1.x, aby = T1.y, abz = T1.z;
    float acx = T2.x, acy = T2.y, acz = T2.z;
    float nab = T0.w;   // ab.ab
    float g   = T1.w;   // ab.ac
    float nac = T2.w;   // ac.ac

    float apx = px - T0.x, apy = py - T0.y, apz = pz - T0.z;
    float bpx = apx - abx, bpy = apy - aby, bpz = apz - abz;   // p - b
    float cpx = apx - acx, cpy = apy - acy, cpz = apz - acz;   // p - c

    float d1 = dot3(abx, aby, abz, apx, apy, apz);
    float d2 = dot3(acx, acy, acz, apx, apy, apz);
    float d3 = d1 - nab;   // ab.(p-b)
    float d4 = d2 - g;     // ac.(p-b)
    float d5 = d1 - g;     // ab.(p-c)
    float d6 = d2 - nac;   // ac.(p-c)

    float va = fmaf(d3, d6, -d5 * d4);
    float vb = fmaf(d5, d2, -d1 * d6);
    float vc = fmaf(d1, d4, -d3 * d2);

    float e13 = d1 - d3;
    float den_ab = (fabsf(e13) > EPS_F) ? e13 : 1.0f;
    float t_ab = clamp01(d1 * frcp(den_ab));
    float e26 = d2 - d6;
    float den_ac = (fabsf(e26) > EPS_F) ? e26 : 1.0f;
    float t_ac = clamp01(d2 * frcp(den_ac));
    float e43 = d4 - d3;
    float e56 = d5 - d6;
    float bcr = e43 + e56;
    float den_bc = (fabsf(bcr) > EPS_F) ? bcr : 1.0f;
    float t_bc = clamp01(e43 * frcp(den_bc));
    float inr = (va + vb) + vc;
    float den_in = (fabsf(inr) > EPS_F) ? inr : 1.0f;
    float rin = frcp(den_in);

    float v = vb * rin, w = vc * rin;
    // apply selects innermost-first so outermost condition wins (matches nested where)
    bool cond_bc = (va <= 0.0f) && (e43 >= 0.0f) && (e56 >= 0.0f);
    v = cond_bc ? (1.0f - t_bc) : v;  w = cond_bc ? t_bc : w;
    bool cond_ac = (vb <= 0.0f) && (d2 >= 0.0f) && (d6 <= 0.0f);
    v = cond_ac ? 0.0f : v;           w = cond_ac ? t_ac : w;
    bool cond_ab = (vc <= 0.0f) && (d1 >= 0.0f) && (d3 <= 0.0f);
    v = cond_ab ? t_ab : v;           w = cond_ab ? 0.0f : w;
    bool cond_c  = (d6 >= 0.0f) && (d5 <= d6);
    v = cond_c ? 0.0f : v;            w = cond_c ? 1.0f : w;
    bool cond_b  = (d3 >= 0.0f) && (d4 <= d3);
    v = cond_b ? 1.0f : v;            w = cond_b ? 0.0f : w;
    bool cond_a  = (d1 <= 0.0f) && (d2 <= 0.0f);
    v = cond_a ? 0.0f : v;            w = cond_a ? 0.0f : w;

    float clx = fmaf(v, abx, fmaf(w, acx, T0.x));
    float cly = fmaf(v, aby, fmaf(w, acy, T0.y));
    float clz = fmaf(v, abz, fmaf(w, acz, T0.z));
    float dx = px - clx, dy = py - cly, dz = pz - clz;
    float sq = fmaf(dx, dx, fmaf(dy, dy, dz * dz));
    minsq = fminf(minsq, sq);

    // ----- winding: ra = -ap, rb = -bp, rc = -cp -----
    float la = fsqrt(fmaxf(dot3(apx, apy, apz, apx, apy, apz), EPS_F));
    float lb = fsqrt(fmaxf(dot3(bpx, bpy, bpz, bpx, bpy, bpz), EPS_F));
    float lc = fsqrt(fmaxf(dot3(cpx, cpy, cpz, cpx, cpy, cpz), EPS_F));
    // cross(rb, rc) == cross(bp, cp); det = ra . (rb x rc) = -ap . (bp x cp)
    float crx = fmaf(bpy, cpz, -bpz * cpy);
    float cry = fmaf(bpz, cpx, -bpx * cpz);
    float crz = fmaf(bpx, cpy, -bpy * cpx);
    float det = -dot3(apx, apy, apz, crx, cry, crz);
    float dab = dot3(apx, apy, apz, bpx, bpy, bpz);   // ra.rb
    float dbc = dot3(bpx, bpy, bpz, cpx, cpy, cpz);   // rb.rc
    float dca = dot3(cpx, cpy, cpz, apx, apy, apz);   // rc.ra
    float denom = fmaf(la * lb, lc, fmaf(dab, lc, fmaf(dbc, la, dca * lb)));
    wsum += fast_atan2f(det, denom);
}

// ---------- kernel 1: gather + precompute per-triangle derived data ----------
__global__ __launch_bounds__(256) void gather_tris(const float* __restrict__ verts,
                                                   const int* __restrict__ idx,
                                                   float4* __restrict__ tris, int ntri) {
    int t = blockIdx.x * blockDim.x + threadIdx.x;
    if (t >= ntri) return;
    int i0 = idx[3 * t + 0], i1 = idx[3 * t + 1], i2 = idx[3 * t + 2];
    const float* a = verts + 3 * i0;
    const float* b = verts + 3 * i1;
    const float* c = verts + 3 * i2;
    float ax = a[0], ay = a[1], az = a[2];
    float abx = b[0] - ax, aby = b[1] - ay, abz = b[2] - az;
    float acx = c[0] - ax, acy = c[1] - ay, acz = c[2] - az;
    float nab = dot3(abx, aby, abz, abx, aby, abz);
    float gg  = dot3(abx, aby, abz, acx, acy, acz);
    float nac = dot3(acx, acy, acz, acx, acy, acz);
    tris[3 * t + 0] = make_float4(ax, ay, az, nab);
    tris[3 * t + 1] = make_float4(abx, aby, abz, gg);
    tris[3 * t + 2] = make_float4(acx, acy, acz, nac);
}

// ---------- async (or sync-fallback) tile loader ----------
__device__ __forceinline__ void load_tile(float4* dst, const float4* src,
                                          int base4, int total4, int tid) {
#pragma unroll
    for (int k = 0; k < COPIES; ++k) {
        int i  = k * THREADS + tid;
        int gi = base4 + i;
        if (gi < total4) {
#if HAS_ASYNC_LDS
            __builtin_amdgcn_global_load_async_to_lds_b128(
                (gptr_v4i)(src + gi),
                (lptr_v4i)(dst + i),
                0, 0);
#else
            dst[i] = src[gi];
#endif
        }
    }
}

// ---------- kernel 2: main SDF, triangles double-buffered through LDS ----------
__global__ __launch_bounds__(THREADS) void sdf_main(const float* __restrict__ pts,
                                                    const float4* __restrict__ tris,
                                                    float* __restrict__ out,
                                                    int npts, int ntri) {
    __shared__ float4 smem[2][F4_PER_TILE];
    int tid = threadIdx.x;
    int pi  = blockIdx.x * THREADS + tid;

    float px = 0.0f, py = 0.0f, pz = 0.0f;
    if (pi < npts) { px = pts[3 * pi]; py = pts[3 * pi + 1]; pz = pts[3 * pi + 2]; }

    float minsq = 3.4e38f;
    float wsum  = 0.0f;

    int ntiles = (ntri + TILE - 1) / TILE;
    int total4 = ntri * 3;

    load_tile(smem[0], tris, 0, total4, tid);

    for (int t = 0; t < ntiles; ++t) {
        int buf = t & 1;
        if (t + 1 < ntiles) {
            load_tile(smem[buf ^ 1], tris, (t + 1) * F4_PER_TILE, total4, tid);
            WAIT_ASYNC(COPIES);   // tile t's copies (issued earlier, in-order) complete
        } else {
            WAIT_ASYNC(0);
        }
        __syncthreads();

        const float4* tb = smem[buf];
        int cnt = min(TILE, ntri - t * TILE);
        for (int j = 0; j < cnt; ++j) {
            float4 T0 = tb[3 * j + 0];
            float4 T1 = tb[3 * j + 1];
            float4 T2 = tb[3 * j + 2];
            pair_accum(px, py, pz, T0, T1, T2, minsq, wsum);
        }
        __syncthreads();
    }

    if (pi < npts) {
        float dist = fsqrt(fmaxf(minsq, EPS_F));
        dist = fminf(dist, MAX_DIST_F);
        float wind = wsum * INV_TWO_PI;             // (sum 2*atan2)/(4*pi)
        float sgn  = (fabsf(wind) > 0.5f) ? -1.0f : 1.0f;
        out[pi] = sgn * dist;
    }
}

// ---------- fallback if workspace is too small: gather tiles in-kernel ----------
__global__ __launch_bounds__(THREADS) void sdf_direct(const float* __restrict__ pts,
                                                      const float* __restrict__ verts,
                                                      const int* __restrict__ idx,
                                                      float* __restrict__ out,
                                                      int npts, int ntri) {
    __shared__ float4 smem[F4_PER_TILE];
    int tid = threadIdx.x;
    int pi  = blockIdx.x * THREADS + tid;

    float px = 0.0f, py = 0.0f, pz = 0.0f;
    if (pi < npts) { px = pts[3 * pi]; py = pts[3 * pi + 1]; pz = pts[3 * pi + 2]; }

    float minsq = 3.4e38f;
    float wsum  = 0.0f;

    int ntiles = (ntri + TILE - 1) / TILE;
    for (int t = 0; t < ntiles; ++t) {
        int base = t * TILE;
        for (int i = tid; i < TILE; i += THREADS) {
            int tr = base + i;
            if (tr < ntri) {
                int i0 = idx[3 * tr + 0], i1 = idx[3 * tr + 1], i2 = idx[3 * tr + 2];
                const float* a = verts + 3 * i0;
                const float* b = verts + 3 * i1;
                const float* c = verts + 3 * i2;
                float ax = a[0], ay = a[1], az = a[2];
                float abx = b[0] - ax, aby = b[1] - ay, abz = b[2] - az;
                float acx = c[0] - ax, acy = c[1] - ay, acz = c[2] - az;
                float nab = dot3(abx, aby, abz, abx, aby, abz);
                float gg  = dot3(abx, aby, abz, acx, acy, acz);
                float nac = dot3(acx, acy, acz, acx, acy, acz);
                smem[3 * i + 0] = make_float4(ax, ay, az, nab);
                smem[3 * i + 1] = make_float4(abx, aby, abz, gg);
                smem[3 * i + 2] = make_float4(acx, acy, acz, nac);
            }
        }
        __syncthreads();
        int cnt = min(TILE, ntri - base);
        for (int j = 0; j < cnt; ++j)
            pair_accum(px, py, pz, smem[3 * j], smem[3 * j + 1], smem[3 * j + 2], minsq, wsum);
        __syncthreads();
    }

    if (pi < npts) {
        float dist = fsqrt(fmaxf(minsq, EPS_F));
        dist = fminf(dist, MAX_DIST_F);
        float wind = wsum * INV_TWO_PI;
        float sgn  = (fabsf(wind) > 0.5f) ? -1.0f : 1.0f;
        out[pi] = sgn * dist;
    }
}

extern "C" void kernel_launch(void* const* d_in, const int* in_sizes, int n_in,
                              void* d_out, int out_size, void* d_ws, size_t ws_size,
                              hipStream_t stream) {
    const float* verts = (const float*)d_in[0];
    const int*   idx   = (const int*)d_in[1];
    const float* pts   = (const float*)d_in[2];
    float*       out   = (float*)d_out;

    int ntri = in_sizes[1] / 3;
    int npts = in_sizes[2] / 3;

    size_t need = (size_t)ntri * 3 * sizeof(float4);
    int pblocks = (npts + THREADS - 1) / THREADS;

    if (ws_size >= need) {
        float4* tris = (float4*)d_ws;
        int gblocks = (ntri + 255) / 256;
        gather_tris<<<gblocks, 256, 0, stream>>>(verts, idx, tris, ntri);
        sdf_main<<<pblocks, THREADS, 0, stream>>>(pts, tris, out, npts, ntri);
    } else {
        sdf_direct<<<pblocks, THREADS, 0, stream>>>(pts, verts, idx, out, npts, ntri);
    }
}